// DifferentiableMarchingCubes_59631325938494
// MI455X (gfx1250) — compile-verified
//
#include <hip/hip_runtime.h>

typedef float v4f __attribute__((ext_vector_type(4)));

// RES=192 SDF grid; edges per axis block = 191*192*192
#define RR 192
#define RM 191
#define NE (RM * RR * RR)            // 7,041,024 edges per axis
#define STEPF (2.0f / 191.0f)
#define THREADS 256
#define BLOCKS_PER_AXIS (NE / 4 / THREADS)   // 6876 exactly

// One thread = 4 consecutive edges of one axis block.
// Output layout (floats): [axis0 verts | axis1 verts | axis2 verts | axis0 mask | axis1 mask | axis2 mask]
template <int AXIS>
__device__ __forceinline__ void mc_axis_body(int blk, const float* __restrict__ sdf,
                                             float* __restrict__ out) {
  const int e0 = (blk * THREADS + (int)threadIdx.x) * 4;

  float a[4], b[4];
  int ii[4], jj[4], kk[4], nn[4];

  if (AXIS == 0) {
    // v0 flat index == e ; v1 = e + 192*192. Contiguous, 16B aligned.
    const v4f va = *(const v4f*)(sdf + e0);
    const v4f vb = *(const v4f*)(sdf + e0 + RR * RR);
    const int i = e0 / (RR * RR);
    const int j = (e0 / RR) % RR;
    const int k0 = e0 % RR;
#pragma unroll
    for (int u = 0; u < 4; ++u) {
      a[u] = va[u]; b[u] = vb[u];
      ii[u] = i; jj[u] = j; kk[u] = k0 + u; nn[u] = i;
    }
    {   // run the streaming SDF read ahead of the wavefront (global_prefetch_b8)
      int pf = e0 + 64 * RR; pf = pf < (RR * RR * RR - 4) ? pf : (RR * RR * RR - 4);
      __builtin_prefetch(sdf + pf, 0, 1);
    }
  } else if (AXIS == 1) {
    // v0 = e + (e/(191*192))*192 ; v1 = v0 + 192. Contiguous, 16B aligned.
    const int i = e0 / (RM * RR);
    const int s = e0 + i * RR;
    const v4f va = *(const v4f*)(sdf + s);
    const v4f vb = *(const v4f*)(sdf + s + RR);
    const int j = (e0 / RR) % RM;
    const int k0 = e0 % RR;
#pragma unroll
    for (int u = 0; u < 4; ++u) {
      a[u] = va[u]; b[u] = vb[u];
      ii[u] = i; jj[u] = j; kk[u] = k0 + u; nn[u] = j;
    }
    {
      int pf = s + 64 * RR; pf = pf < (RR * RR * RR - 4) ? pf : (RR * RR * RR - 4);
      __builtin_prefetch(sdf + pf, 0, 1);
    }
  } else {
    // Rows of length 191: v0 = e + e/191 ; v1 = v0 + 1. Scalar loads (L1/L2 hits).
#pragma unroll
    for (int u = 0; u < 4; ++u) {
      const int e = e0 + u;
      const int q = e / RM;          // = i*192 + j
      const int k = e - q * RM;
      const int s = e + q;           // = q*192 + k
      a[u] = sdf[s]; b[u] = sdf[s + 1];
      ii[u] = q / RR; jj[u] = q % RR; kk[u] = k; nn[u] = k;
    }
  }

  float vx[4], vy[4], vz[4], mm[4];
#pragma unroll
  for (int u = 0; u < 4; ++u) {
    const float v0 = a[u], v1 = b[u];
    const bool cross = (v0 < 0.0f) != (v1 < 0.0f);
    const float t = fminf(fmaxf((0.0f - v0) / (v1 - v0 + 1e-8f), 0.0f), 1.0f);
    float x = -1.0f + (float)ii[u] * STEPF;
    float y = -1.0f + (float)jj[u] * STEPF;
    float z = -1.0f + (float)kk[u] * STEPF;
    const float interp = (-1.0f + (float)nn[u] * STEPF) + t * STEPF;  // p0a + t*(p1a-p0a)
    if (AXIS == 0) x = interp;
    else if (AXIS == 1) y = interp;
    else z = interp;
    vx[u] = cross ? x : 0.0f;
    vy[u] = cross ? y : 0.0f;
    vz[u] = cross ? z : 0.0f;
    mm[u] = cross ? 1.0f : 0.0f;
  }

  // 4 edges -> 48B of vertex data: three aligned b128 NT stores + one b128 NT mask store.
  // Output (338 MB) is write-once and larger than the 192 MB L2: NT keeps the
  // L2 reserved for the 28 MB SDF that all three axis passes re-read.
  float* vp = out + AXIS * (3 * NE) + e0 * 3;
  float* mp = out + 9 * NE + AXIS * NE + e0;
  const v4f q0 = {vx[0], vy[0], vz[0], vx[1]};
  const v4f q1 = {vy[1], vz[1], vx[2], vy[2]};
  const v4f q2 = {vz[2], vx[3], vy[3], vz[3]};
  const v4f qm = {mm[0], mm[1], mm[2], mm[3]};
  __builtin_nontemporal_store(q0, (v4f*)vp + 0);
  __builtin_nontemporal_store(q1, (v4f*)vp + 1);
  __builtin_nontemporal_store(q2, (v4f*)vp + 2);
  __builtin_nontemporal_store(qm, (v4f*)mp);
}

// Single fused launch: axis selected block-uniformly (no divergence), all three
// axis streams in flight together -> one dispatch ramp instead of three, and
// axis-2's cache-hit read phases interleave with the store-bound axis-0/1 blocks.
__global__ __launch_bounds__(THREADS) void mc_edges_fused(const float* __restrict__ sdf,
                                                          float* __restrict__ out) {
  const int axis = (int)blockIdx.x / BLOCKS_PER_AXIS;
  const int blk  = (int)blockIdx.x % BLOCKS_PER_AXIS;
  if (axis == 0)      mc_axis_body<0>(blk, sdf, out);
  else if (axis == 1) mc_axis_body<1>(blk, sdf, out);
  else                mc_axis_body<2>(blk, sdf, out);
}

extern "C" void kernel_launch(void* const* d_in, const int* in_sizes, int n_in,
                              void* d_out, int out_size, void* d_ws, size_t ws_size,
                              hipStream_t stream) {
  (void)in_sizes; (void)n_in; (void)out_size; (void)d_ws; (void)ws_size;
  const float* sdf = (const float*)d_in[0];
  float* out = (float*)d_out;

  mc_edges_fused<<<3 * BLOCKS_PER_AXIS, THREADS, 0, stream>>>(sdf, out);
}